// ProdLayer_36764920054638
// MI455X (gfx1250) — compile-verified
//
#include <hip/hip_runtime.h>
#include <stdint.h>

// ProdLayer forward: element_mars[nids] = node_mars[cids].sum(axis=1)
// Pure gather/sum/scatter, HBM-bound (~1.28 GB traffic -> ~55us at 23.3 TB/s).
// Strategy: one wave32 per node; gather the 4 child rows (4 x 512B) with
// global_load_async_to_lds_b128 (per-lane async gather engine, ASYNCcnt),
// double-buffered in LDS so the next node's gather overlaps the current sum.
// Output is write-once -> non-temporal stores to preserve L2 for gather reuse.

typedef __attribute__((ext_vector_type(4))) float f32x4;
typedef __attribute__((ext_vector_type(4))) int   i32x4;

#define WAVES_PER_BLOCK 8
#define THREADS (WAVES_PER_BLOCK * 32)
#define B_COLS 128
#define ROW_BYTES (B_COLS * 4)          // 512 B per row
#define NODE_BYTES (4 * ROW_BYTES)      // 2 KB per node (4 children)
#define NODE_FLOATS (4 * B_COLS)        // 512 floats

// Issue 4 async b128 gathers: one full 512B row per instruction (32 lanes x 16B).
__device__ __forceinline__ void issue_gather(uint64_t nm_base,
                                             const i32x4* __restrict__ cids4,
                                             int n, uint32_t lds_buf,
                                             uint32_t lane16) {
    i32x4 c = cids4[n];   // 16B broadcast load (all lanes same address)
    uint64_t a0 = nm_base + ((uint64_t)(uint32_t)c.x << 9) + lane16;
    uint64_t a1 = nm_base + ((uint64_t)(uint32_t)c.y << 9) + lane16;
    uint64_t a2 = nm_base + ((uint64_t)(uint32_t)c.z << 9) + lane16;
    uint64_t a3 = nm_base + ((uint64_t)(uint32_t)c.w << 9) + lane16;
    uint32_t l0 = lds_buf + lane16;
    uint32_t l1 = l0 + ROW_BYTES;
    uint32_t l2 = l0 + 2 * ROW_BYTES;
    uint32_t l3 = l0 + 3 * ROW_BYTES;
    asm volatile("global_load_async_to_lds_b128 %0, %1, off" :: "v"(l0), "v"(a0) : "memory");
    asm volatile("global_load_async_to_lds_b128 %0, %1, off" :: "v"(l1), "v"(a1) : "memory");
    asm volatile("global_load_async_to_lds_b128 %0, %1, off" :: "v"(l2), "v"(a2) : "memory");
    asm volatile("global_load_async_to_lds_b128 %0, %1, off" :: "v"(l3), "v"(a3) : "memory");
}

__global__ __launch_bounds__(THREADS)
void prod_layer_async_kernel(const float* __restrict__ node_mars,
                             const float* __restrict__ element_mars,
                             const int*   __restrict__ nids,
                             const i32x4* __restrict__ cids4,
                             float*       __restrict__ out,
                             int n_prod) {
    __shared__ float smem[WAVES_PER_BLOCK * 2 * NODE_FLOATS];  // 32 KB

    const int lane  = threadIdx.x & 31;
    const int wib   = threadIdx.x >> 5;
    const int wave  = blockIdx.x * WAVES_PER_BLOCK + wib;
    const int nwav  = gridDim.x * WAVES_PER_BLOCK;
    const uint32_t lane16 = (uint32_t)(lane << 4);

    // Output row 0 = element_mars row 0 (reserved zero/padding row).
    if (wave == 0) {
        f32x4 z = ((const f32x4*)element_mars)[lane];
        __builtin_nontemporal_store(z, (f32x4*)out + lane);
    }

    const uint64_t nm_base = (uint64_t)(uintptr_t)node_mars;
    // LDS aperture: flat addr low 32 bits == LDS byte offset (ISA 10.2).
    const uint32_t lds_wave =
        (uint32_t)(uintptr_t)&smem[0] + (uint32_t)(wib * 2 * NODE_BYTES);

    int n = wave;
    int buf = 0;
    if (n < n_prod)
        issue_gather(nm_base, cids4, n, lds_wave, lane16);

    while (n < n_prod) {
        const int n_next = n + nwav;
        if (n_next < n_prod) {
            issue_gather(nm_base, cids4, n_next, lds_wave + (uint32_t)((buf ^ 1) * NODE_BYTES), lane16);
            // 4 just issued may be outstanding; previous node's 4 must be done
            // (async loads complete in order).
            asm volatile("s_wait_asynccnt 0x4" ::: "memory");
        } else {
            asm volatile("s_wait_asynccnt 0x0" ::: "memory");
        }

        // Sum the 4 gathered rows for node n from LDS (ds_load_b128 x4).
        const float* bp = &smem[wib * 2 * NODE_FLOATS + buf * NODE_FLOATS + lane * 4];
        f32x4 r0 = *(const f32x4*)(bp);
        f32x4 r1 = *(const f32x4*)(bp + B_COLS);
        f32x4 r2 = *(const f32x4*)(bp + 2 * B_COLS);
        f32x4 r3 = *(const f32x4*)(bp + 3 * B_COLS);
        f32x4 s = (r0 + r1) + (r2 + r3);

        const int row = nids[n];  // uniform per wave
        __builtin_nontemporal_store(s, (f32x4*)(out + (size_t)row * B_COLS) + lane);

        n = n_next;
        buf ^= 1;
    }
}

extern "C" void kernel_launch(void* const* d_in, const int* in_sizes, int n_in,
                              void* d_out, int out_size, void* d_ws, size_t ws_size,
                              hipStream_t stream) {
    const float* node_mars    = (const float*)d_in[0];
    const float* element_mars = (const float*)d_in[1];
    const int*   nids         = (const int*)d_in[2];
    const i32x4* cids4        = (const i32x4*)d_in[3];
    float*       out          = (float*)d_out;

    const int n_prod = in_sizes[2];  // 500,000

    int blocks = (n_prod + WAVES_PER_BLOCK - 1) / WAVES_PER_BLOCK;
    if (blocks > 2048) blocks = 2048;  // 16384 waves, ~31 nodes/wave pipeline depth
    if (blocks < 1) blocks = 1;

    prod_layer_async_kernel<<<blocks, THREADS, 0, stream>>>(
        node_mars, element_mars, nids, cids4, out, n_prod);
}